// STFT_58987080843818
// MI455X (gfx1250) — compile-verified
//
#include <hip/hip_runtime.h>

typedef __attribute__((ext_vector_type(2))) float v2f;
typedef __attribute__((ext_vector_type(4))) float f4;
typedef __attribute__((ext_vector_type(8))) float v8f;

#define FILT 1024
#define HOP  512
#define NCH  1026      // 2 * CUTOFF channels
#define NFRM 1027      // frames per batch
#define NS   524288    // samples per batch
#define KTOT2 2052     // 2 * NCH (inverse GEMM K)
#define LDA  34        // LDS row stride (even -> 8B aligned v2f frags, gcd(34,64)=2)

// ---------------------------------------------------------------------------
// Stage 1: ft[b,c,t] = sum_k FB[c,k] * x[b, t*512 + k - 1024]   (zero padded)
// GEMM: A = FB (1026 x 1024), B = frames (1024 x 1027), per batch.
// Block: 128(M) x 64(N) tile, 8 waves, wave = 16(M) x 64(N) -> 4 accumulators.
// ---------------------------------------------------------------------------
__global__ __launch_bounds__(256) void stft_fwd(const float* __restrict__ x,
                                                const float* __restrict__ fb,
                                                float* __restrict__ ft)
{
    __shared__ float As[128 * LDA];   // A tile: 128 M x 32 K
    __shared__ float Bs[64  * LDA];   // B tile: 64  N x 32 K (N-major)
    const int tid  = threadIdx.x;
    const int lane = tid & 31;
    const int wv   = tid >> 5;
    const int t0   = blockIdx.x * 64;    // frame tile
    const int m0   = blockIdx.y * 128;   // channel tile
    const int b    = blockIdx.z;
    const float* xb = x + (size_t)b * NS;

    v8f acc0 = {}, acc1 = {}, acc2 = {}, acc3 = {};

    for (int kc = 0; kc < FILT; kc += 32) {
        // ---- load A tile (coalesced along k) ----
#pragma unroll
        for (int i = 0; i < 4; ++i) {
            int g  = tid + i * 256;        // 0..1023 float4 slots
            int ml = g >> 3;               // 0..127
            int kk = (g & 7) * 4;          // 0..28
            f4 v = {};
            int c = m0 + ml;
            if (c < NCH) v = *(const f4*)(fb + (size_t)c * FILT + kc + kk);
            float* dst = &As[ml * LDA + kk];
            dst[0] = v.x; dst[1] = v.y; dst[2] = v.z; dst[3] = v.w;
        }
        // ---- load B tile: frames gathered from x (16B aligned, no straddle) ----
#pragma unroll
        for (int i = 0; i < 2; ++i) {
            int g  = tid + i * 256;        // 0..511 float4 slots
            int tl = g & 63;               // 0..63
            int kk = (g >> 6) * 4;         // 0..28
            int t  = t0 + tl;
            long n0 = (long)t * HOP + kc + kk - FILT;
            f4 v = {};
            if (t < NFRM && n0 >= 0 && n0 + 3 < NS)
                v = *(const f4*)(xb + n0);
            float* dst = &Bs[tl * LDA + kk];
            dst[0] = v.x; dst[1] = v.y; dst[2] = v.z; dst[3] = v.w;
        }
        __syncthreads();
        // ---- 8 WMMA K-steps of 4 ----
#pragma unroll
        for (int ks = 0; ks < 8; ++ks) {
            int kb = ks * 4 + 2 * (lane >> 4);                 // K = kb + j (j=0,1)
            v2f a  = *(const v2f*)&As[(wv * 16 + (lane & 15)) * LDA + kb];
            v2f b0 = *(const v2f*)&Bs[( 0 + (lane & 15)) * LDA + kb];
            v2f b1 = *(const v2f*)&Bs[(16 + (lane & 15)) * LDA + kb];
            v2f b2 = *(const v2f*)&Bs[(32 + (lane & 15)) * LDA + kb];
            v2f b3 = *(const v2f*)&Bs[(48 + (lane & 15)) * LDA + kb];
            acc0 = __builtin_amdgcn_wmma_f32_16x16x4_f32(false, a, false, b0, (short)0, acc0, false, false);
            acc1 = __builtin_amdgcn_wmma_f32_16x16x4_f32(false, a, false, b1, (short)0, acc1, false, false);
            acc2 = __builtin_amdgcn_wmma_f32_16x16x4_f32(false, a, false, b2, (short)0, acc2, false, false);
            acc3 = __builtin_amdgcn_wmma_f32_16x16x4_f32(false, a, false, b3, (short)0, acc3, false, false);
        }
        __syncthreads();
    }
    // ---- store D (C/D layout: reg r -> M = r + 8*(lane>=16), N = lane&15) ----
    int mrow  = m0 + wv * 16 + ((lane >> 4) << 3);
    int tbase = t0 + (lane & 15);
#pragma unroll
    for (int rr = 0; rr < 8; ++rr) {
        int c = mrow + rr;
        if (c < NCH) {
            size_t base = ((size_t)b * NCH + c) * NFRM;
            if (tbase      < NFRM) ft[base + tbase     ] = acc0[rr];
            if (tbase + 16 < NFRM) ft[base + tbase + 16] = acc1[rr];
            if (tbase + 32 < NFRM) ft[base + tbase + 32] = acc2[rr];
            if (tbase + 48 < NFRM) ft[base + tbase + 48] = acc3[rr];
        }
    }
}

// ---------------------------------------------------------------------------
// Stage 2: out[b, 512j + r] = sum_c ft[b,c,j+1]*IB[c,r+512] + ft[b,c,j+2]*IB[c,r]
// GEMM per batch: M = 1024 (j), N = 512 (r), K = 2052 (c' = c or c+1026).
// ---------------------------------------------------------------------------
__global__ __launch_bounds__(256) void stft_inv(const float* __restrict__ ft,
                                                const float* __restrict__ ib,
                                                float* __restrict__ out)
{
    __shared__ float As[128 * LDA];   // A tile: 128 j x 32 k'
    __shared__ float Bs[64  * LDA];   // B tile: 64  r x 32 k'
    const int tid  = threadIdx.x;
    const int lane = tid & 31;
    const int wv   = tid >> 5;
    const int r0   = blockIdx.x * 64;    // output offset within hop-chunk
    const int j0   = blockIdx.y * 128;   // hop-chunk tile
    const int b    = blockIdx.z;
    const float* ftb = ft + (size_t)b * NCH * NFRM;

    v8f acc0 = {}, acc1 = {}, acc2 = {}, acc3 = {};

    for (int kc = 0; kc < KTOT2; kc += 32) {   // 65 chunks, last zero-padded
        // ---- load A tile: ft transposed gather, contiguous along j(t) ----
#pragma unroll
        for (int i = 0; i < 4; ++i) {
            int g  = tid + i * 256;       // 0..1023
            int kl = g >> 5;              // 0..31
            int jl = (g & 31) * 4;        // 0..124
            f4 v = {};
            int cp = kc + kl;
            if (cp < KTOT2) {
                int s  = (cp >= NCH) ? 1 : 0;
                int cc = cp - (s ? NCH : 0);
                const float* src = ftb + (size_t)cc * NFRM + (j0 + jl + 1 + s);
                v.x = src[0]; v.y = src[1]; v.z = src[2]; v.w = src[3];
            }
            float* dst = &As[jl * LDA + kl];
            dst[0] = v.x; dst[LDA] = v.y; dst[2 * LDA] = v.z; dst[3 * LDA] = v.w;
        }
        // ---- load B tile: W[k',r] = IB[cc, r + (s ? 0 : 512)] ----
#pragma unroll
        for (int i = 0; i < 2; ++i) {
            int g  = tid + i * 256;       // 0..511
            int kl = g >> 4;              // 0..31
            int rl = (g & 15) * 4;        // 0..60
            f4 v = {};
            int cp = kc + kl;
            if (cp < KTOT2) {
                int s  = (cp >= NCH) ? 1 : 0;
                int cc = cp - (s ? NCH : 0);
                v = *(const f4*)(ib + (size_t)cc * FILT + r0 + rl + (s ? 0 : HOP));
            }
            float* dst = &Bs[rl * LDA + kl];
            dst[0] = v.x; dst[LDA] = v.y; dst[2 * LDA] = v.z; dst[3 * LDA] = v.w;
        }
        __syncthreads();
#pragma unroll
        for (int ks = 0; ks < 8; ++ks) {
            int kb = ks * 4 + 2 * (lane >> 4);
            v2f a  = *(const v2f*)&As[(wv * 16 + (lane & 15)) * LDA + kb];
            v2f b0 = *(const v2f*)&Bs[( 0 + (lane & 15)) * LDA + kb];
            v2f b1 = *(const v2f*)&Bs[(16 + (lane & 15)) * LDA + kb];
            v2f b2 = *(const v2f*)&Bs[(32 + (lane & 15)) * LDA + kb];
            v2f b3 = *(const v2f*)&Bs[(48 + (lane & 15)) * LDA + kb];
            acc0 = __builtin_amdgcn_wmma_f32_16x16x4_f32(false, a, false, b0, (short)0, acc0, false, false);
            acc1 = __builtin_amdgcn_wmma_f32_16x16x4_f32(false, a, false, b1, (short)0, acc1, false, false);
            acc2 = __builtin_amdgcn_wmma_f32_16x16x4_f32(false, a, false, b2, (short)0, acc2, false, false);
            acc3 = __builtin_amdgcn_wmma_f32_16x16x4_f32(false, a, false, b3, (short)0, acc3, false, false);
        }
        __syncthreads();
    }
    // ---- store: out[b, j*512 + r], all indices in range by construction ----
    float* ob = out + (size_t)b * NS;
    int jrow = j0 + wv * 16 + ((lane >> 4) << 3);
    int rcol = r0 + (lane & 15);
#pragma unroll
    for (int rr = 0; rr < 8; ++rr) {
        size_t base = (size_t)(jrow + rr) * HOP + rcol;
        ob[base     ] = acc0[rr];
        ob[base + 16] = acc1[rr];
        ob[base + 32] = acc2[rr];
        ob[base + 48] = acc3[rr];
    }
}

extern "C" void kernel_launch(void* const* d_in, const int* in_sizes, int n_in,
                              void* d_out, int out_size, void* d_ws, size_t ws_size,
                              hipStream_t stream) {
    const float* x  = (const float*)d_in[0];   // (32, 524288)
    const float* fb = (const float*)d_in[1];   // (1026, 1, 1024)
    const float* ib = (const float*)d_in[2];   // (1026, 1, 1024)
    float* out = (float*)d_out;                // (32, 1, 524288)
    float* ft  = (float*)d_ws;                 // (32, 1026, 1027) fp32, ~135 MB scratch

    (void)in_sizes; (void)n_in; (void)out_size; (void)ws_size;

    dim3 g1((NFRM + 63) / 64, (NCH + 127) / 128, 32);   // 17 x 9 x 32
    stft_fwd<<<g1, dim3(256), 0, stream>>>(x, fb, ft);

    dim3 g2(512 / 64, 1024 / 128, 32);                  // 8 x 8 x 32
    stft_inv<<<g2, dim3(256), 0, stream>>>(ft, ib, out);
}